// MultiLabelContrastiveLoss_40175124087458
// MI455X (gfx1250) — compile-verified
//
#include <hip/hip_runtime.h>
#include <stdint.h>

#define B_N 8192
#define D_K 128
#define N_ASPECT 12
// exp(sim / 0.07) == exp2(sim * log2(e)/0.07)
#define EXP2_SCALE 20.6098649276958f   // 1.4426950408889634 / 0.07
#define EPSV 1e-8f

typedef __attribute__((ext_vector_type(2))) float v2f;
typedef __attribute__((ext_vector_type(4))) float v4f;
typedef __attribute__((ext_vector_type(8))) float v8f;

#if __has_builtin(__builtin_amdgcn_exp2f)
#define FAST_EXP2(x) __builtin_amdgcn_exp2f(x)
#else
#define FAST_EXP2(x) exp2f(x)
#endif

// ---------------------------------------------------------------------------
// Kernel 1a: pack one-hot(3) x 12 aspects into a 36-bit mask per row, and
// zero the per-row pos/all accumulators (ws is poisoned, not zeroed).
__global__ __launch_bounds__(256) void pack_labels_zero(
    const int* __restrict__ labels,
    unsigned long long* __restrict__ bits,
    float* __restrict__ pos, float* __restrict__ alls) {
  int i = blockIdx.x * blockDim.x + threadIdx.x;
  if (i < B_N) {
    unsigned long long b = 0ull;
#pragma unroll
    for (int a = 0; a < N_ASPECT; ++a) {
      int lab = labels[i * N_ASPECT + a];
      b |= 1ull << (3 * a + lab);
    }
    bits[i] = b;
    pos[i] = 0.0f;
    alls[i] = 0.0f;
  }
}

// ---------------------------------------------------------------------------
// Kernel 1b: swizzle E so WMMA fragments are lane-contiguous.
// For the f32 16x16x4 WMMA, lane half hi needs float pair {4kc+2hi, 4kc+2hi+1}
// per K-chunk kc. Regroup each 128-float row as:
//   Esw[row][hi*64 + 2*kc + t] = E[row][4*kc + 2*hi + t]
// so one b128 load supplies TWO consecutive chunks for a lane.
__global__ __launch_bounds__(256) void swizzle_E(
    const float* __restrict__ E, float* __restrict__ Esw) {
  int t = blockIdx.x * blockDim.x + threadIdx.x;   // B_N * 32 threads
  int row = t >> 5;
  int kc  = t & 31;
  v4f q = *(const v4f*)(E + (size_t)row * D_K + 4 * kc);
  *(v2f*)(Esw + (size_t)row * D_K + 2 * kc)      = q.lo;   // hi=0 half
  *(v2f*)(Esw + (size_t)row * D_K + 64 + 2 * kc) = q.hi;   // hi=1 half
}

// ---------------------------------------------------------------------------
// Kernel 2: per block = one 16-row i-tile; 8 waves each sweep a stripe of
// 16-col j-tiles. f32 WMMA (16x16x4) accumulates sim = E.E^T over K=128.
// 16 b128 loads fill the B tile; sched_group_barrier pins the schedule to
// [VMEM-read group][32x WMMA group] so loads overlap the WMMA chain.
__global__ __launch_bounds__(256)
__attribute__((amdgpu_waves_per_eu(4)))
void sim_loss_rows(
    const float* __restrict__ Esw,
    const unsigned long long* __restrict__ bits,
    float* __restrict__ pos, float* __restrict__ alls) {
  const int i16  = blockIdx.x * 16;
  const int lane = threadIdx.x & 31;
  const int wave = threadIdx.x >> 5;
  const int hi   = lane >> 4;     // which 16-lane half
  const int lm   = lane & 15;

  // Preload A fragments for this i-tile (16 b128 loads; swizzled layout).
  v4f areg[16];
  const float* arow = Esw + (size_t)(i16 + lm) * D_K + hi * 64;
#pragma unroll
  for (int g = 0; g < 16; ++g)
    areg[g] = *(const v4f*)(arow + 4 * g);

  // Label masks for the 8 rows this lane's C registers cover.
  unsigned long long bi[8];
#pragma unroll
  for (int r = 0; r < 8; ++r)
    bi[r] = bits[i16 + r + 8 * hi];

  float posp[8], allp[8];
#pragma unroll
  for (int r = 0; r < 8; ++r) { posp[r] = 0.f; allp[r] = 0.f; }

#pragma unroll 1
  for (int jt = wave; jt < B_N / 16; jt += 8) {
    const int j16 = jt * 16;
    const int jcol = j16 + lm;
    const float* brow = Esw + (size_t)jcol * D_K + hi * 64;

    // Label mask for this lane's column (VMEM read; hides under WMMA chain).
    const unsigned long long bj = bits[jcol];

    // Full B tile for this lane: 16 independent b128 loads (2 chunks each).
    v4f breg[16];
#pragma unroll
    for (int g = 0; g < 16; ++g)
      breg[g] = *(const v4f*)(brow + 4 * g);

    v8f c = {};
#pragma unroll
    for (int g = 0; g < 16; ++g) {
      c = __builtin_amdgcn_wmma_f32_16x16x4_f32(
          false, (v2f)areg[g].lo, false, (v2f)breg[g].lo, (short)0, c,
          false, false);
      c = __builtin_amdgcn_wmma_f32_16x16x4_f32(
          false, (v2f)areg[g].hi, false, (v2f)breg[g].hi, (short)0, c,
          false, false);
    }

    // Pin schedule: issue the load clause first, then the WMMA chain, so
    // s_wait_loadcnt degrades to staged waits (loads return in order).
    __builtin_amdgcn_sched_group_barrier(0x020, 17, 0);  // VMEM reads
    __builtin_amdgcn_sched_group_barrier(0x008, 32, 0);  // WMMA chain

#pragma unroll
    for (int r = 0; r < 8; ++r) {
      const int irow = i16 + r + 8 * hi;
      float e = FAST_EXP2(c[r] * EXP2_SCALE);      // exp(sim / T)
      e = (irow != jcol) ? e : 0.0f;               // off-diagonal mask
      allp[r] += e;
      const int matches = __popcll(bi[r] & bj);    // label_sim * 12
      posp[r] += (matches >= 6) ? e : 0.0f;        // label_sim >= 0.5
    }
  }

  // Reduce each row sum across its 16-lane half, then atomically accumulate.
#pragma unroll
  for (int r = 0; r < 8; ++r) {
    float p = posp[r], a = allp[r];
#pragma unroll
    for (int m = 1; m < 16; m <<= 1) {
      p += __shfl_xor(p, m, 32);
      a += __shfl_xor(a, m, 32);
    }
    if (lm == 0) {
      const int irow = i16 + r + 8 * hi;
      atomicAdd(&pos[irow], p);
      atomicAdd(&alls[irow], a);
    }
  }
}

// ---------------------------------------------------------------------------
// Kernel 3: scalar loss over valid rows.
__global__ __launch_bounds__(256) void finalize_loss(
    const float* __restrict__ pos, const float* __restrict__ alls,
    float* __restrict__ out) {
  __shared__ float ssum[256];
  __shared__ int   scnt[256];
  float lsum = 0.f;
  int   lcnt = 0;
  for (int i = threadIdx.x; i < B_N; i += 256) {
    const float p = pos[i];
    const float a = alls[i];
    if (p > 0.f) {
      lsum += -logf(p / (a + EPSV));
      lcnt += 1;
    }
  }
  ssum[threadIdx.x] = lsum;
  scnt[threadIdx.x] = lcnt;
  __syncthreads();
  for (int s = 128; s > 0; s >>= 1) {
    if (threadIdx.x < s) {
      ssum[threadIdx.x] += ssum[threadIdx.x + s];
      scnt[threadIdx.x] += scnt[threadIdx.x + s];
    }
    __syncthreads();
  }
  if (threadIdx.x == 0)
    out[0] = (scnt[0] > 0) ? ssum[0] / (float)scnt[0] : 0.0f;
}

extern "C" void kernel_launch(void* const* d_in, const int* in_sizes, int n_in,
                              void* d_out, int out_size, void* d_ws, size_t ws_size,
                              hipStream_t stream) {
  const float* E      = (const float*)d_in[0];   // [8192,128] f32, L2-normalized
  const int*   labels = (const int*)d_in[1];     // [8192,12] int32
  float*       out    = (float*)d_out;           // scalar f32

  unsigned long long* bits = (unsigned long long*)d_ws;      // 64 KB
  float* pos  = (float*)(bits + B_N);                        // 32 KB
  float* alls = pos + B_N;                                   // 32 KB
  float* Esw  = alls + B_N;                                  // 4 MB swizzled E

  pack_labels_zero<<<B_N / 256, 256, 0, stream>>>(labels, bits, pos, alls);
  swizzle_E<<<(B_N * 32) / 256, 256, 0, stream>>>(E, Esw);
  sim_loss_rows<<<B_N / 16, 256, 0, stream>>>(Esw, bits, pos, alls);
  finalize_loss<<<1, 256, 0, stream>>>(pos, alls, out);
}